// NestedNERModule_10642928959699
// MI455X (gfx1250) — compile-verified
//
#include <hip/hip_runtime.h>

typedef __attribute__((ext_vector_type(2))) float v2f;
typedef __attribute__((ext_vector_type(8))) float v8f;

#define N_DOCS   64
#define T_TOK    512
#define D_DIM    1024
#define LB       32
#define NUM_TAGS 5
#define K_COLS   (LB * NUM_TAGS)       // 160
#define N_ROWS   (N_DOCS * T_TOK)      // 32768
#define N_SEQ    (N_DOCS * LB)         // 2048
#define IMPOSS   (-10000.0f)

// ---------------------------------------------------------------------------
// K1: spans -> tags scatter (atomicMax; O=0,I=1,B=2,L=3,U=4 so max is exact)
// ---------------------------------------------------------------------------
__global__ void spans_to_tags_kernel(const int* __restrict__ spans,
                                     int* __restrict__ tags, int n_spans) {
    int i = blockIdx.x * blockDim.x + threadIdx.x;
    if (i >= n_spans) return;
    int d = spans[4 * i + 0];
    int l = spans[4 * i + 1];
    int b = spans[4 * i + 2];
    int e = spans[4 * i + 3];
    for (int p = b; p < e; ++p) {
        if (p < 0 || p >= T_TOK) continue;
        int tv;
        if (p == b && p == e - 1)      tv = 4;   // U
        else if (p == b)               tv = 2;   // B
        else if (p == e - 1)           tv = 3;   // L
        else                           tv = 1;   // I
        atomicMax(&tags[((size_t)d * LB + l) * T_TOK + p], tv);
    }
}

// ---------------------------------------------------------------------------
// K2: logits[n,l,t,tag] = embeds[n,t,:] . W[l*5+tag,:] + bias
// fp32 WMMA 16x16x4. Each wave owns one 16-row tile x FIVE 16-col tiles
// (A reused 5x -> 1.2 loads per WMMA). 2 waves/block cover all 160 cols.
// The 5 B tiles sit at constant 64KB strides -> single addr + inst offsets.
// ---------------------------------------------------------------------------
__global__ void __launch_bounds__(64)
gemm_logits_kernel(const float* __restrict__ E, const float* __restrict__ W,
                   const float* __restrict__ bias, float* __restrict__ out) {
    const int lane = threadIdx.x;        // 0..31
    const int wv   = threadIdx.y;        // 0..1 : column half (80 cols each)
    const int mT   = blockIdx.x;         // 0..2047: row tile
    const int mn   = lane & 15;          // M for A, N for B/D
    const int half = lane >> 4;          // selects K pair / M half
    const int colBase = wv * 80;

    const float* aPtr = E + (size_t)(mT * 16 + mn) * D_DIM + 2 * half;
    const float* bPtr = W + (size_t)(colBase + mn) * D_DIM + 2 * half;

    v8f c[5] = {};
#pragma unroll 2
    for (int k = 0; k < D_DIM / 4; ++k) {
        v2f a = *reinterpret_cast<const v2f*>(aPtr + k * 4);
#pragma unroll
        for (int j = 0; j < 5; ++j) {
            // B tile j at constant element offset j*16*D_DIM (64KB stride)
            v2f b = *reinterpret_cast<const v2f*>(bPtr + (size_t)j * 16 * D_DIM + k * 4);
            c[j] = __builtin_amdgcn_wmma_f32_16x16x4_f32(false, a, false, b,
                                                         (short)0, c[j], false, false);
        }
    }

#pragma unroll
    for (int j = 0; j < 5; ++j) {
        const int   col = colBase + j * 16 + mn;   // 0..159
        const float bv  = bias[col];
        const int   l   = col / NUM_TAGS;
        const int   tag = col - l * NUM_TAGS;
#pragma unroll
        for (int i = 0; i < 8; ++i) {
            int row = mT * 16 + i + 8 * half;      // global token row
            int n   = row >> 9;                    // / T_TOK
            int t   = row & (T_TOK - 1);
            out[(((size_t)n * LB + l) * T_TOK + t) * NUM_TAGS + tag] = c[j][i] + bv;
        }
    }
}

// ---------------------------------------------------------------------------
// K3: CRF forward scan — one lane per (sequence, pass). pass 0 = free logits,
// pass 1 = gold-clamped. Clamp is a per-lane select: no control divergence.
// Structural transitions ({O,L,U}->{O,B,U}, {I,B}->{I,L}); exp(-1e4)
// underflows to exactly 0 in f32, identical to the additive reference.
// ---------------------------------------------------------------------------
__device__ __forceinline__ float lse2(float a, float b) {
    float m = fmaxf(a, b);
    return m + __logf(__expf(a - m) + __expf(b - m));
}
__device__ __forceinline__ float lse3(float a, float b, float c) {
    float m = fmaxf(fmaxf(a, b), c);
    return m + __logf(__expf(a - m) + __expf(b - m) + __expf(c - m));
}

__global__ void crf_scan_kernel(const float* __restrict__ logits,
                                const int* __restrict__ tags,
                                const unsigned char* __restrict__ mask,
                                float* __restrict__ z_out) {
    int idx = blockIdx.x * blockDim.x + threadIdx.x;
    if (idx >= 2 * N_SEQ) return;
    const int seq  = idx >> 1;
    const int pass = idx & 1;            // 0 = free, 1 = clamped

    const float* Lg = logits + (size_t)seq * T_TOK * NUM_TAGS;
    const int*   tg = tags   + (size_t)seq * T_TOK;
    const unsigned char* mk = mask + (size_t)(seq >> 5) * T_TOK;  // n = seq/LB

    float a0, a1, a2, a3, a4;
    float lg[5];
#pragma unroll
    for (int j = 0; j < 5; ++j) lg[j] = Lg[j];
    int t0 = tg[0];
    if (pass) {
#pragma unroll
        for (int j = 0; j < 5; ++j) lg[j] = (j == t0) ? lg[j] : IMPOSS;
    }
    // start allows O,B,U
    a0 = lg[0]; a1 = IMPOSS + lg[1]; a2 = lg[2]; a3 = IMPOSS + lg[3]; a4 = lg[4];

    for (int t = 1; t < T_TOK; ++t) {
#pragma unroll
        for (int j = 0; j < 5; ++j) lg[j] = Lg[(size_t)t * NUM_TAGS + j];
        int  tt = tg[t];
        bool mt = mk[t] != 0;
        if (pass) {
#pragma unroll
            for (int j = 0; j < 5; ++j) lg[j] = (j == tt) ? lg[j] : IMPOSS;
        }

        float olu = lse3(a0, a3, a4);    // predecessors of O, B, U
        float ib  = lse2(a1, a2);        // predecessors of I, L
        float n0 = olu + lg[0], n1 = ib + lg[1], n2 = olu + lg[2];
        float n3 = ib + lg[3],  n4 = olu + lg[4];
        if (mt) { a0 = n0; a1 = n1; a2 = n2; a3 = n3; a4 = n4; }
    }

    // end allows O, L, U
    z_out[idx] = lse3(a0, a3, a4);
}

// ---------------------------------------------------------------------------
// K4: per_seq[s] = z[2s] - z[2s+1]; sum + invalid check -> loss scalar
// ---------------------------------------------------------------------------
__global__ void reduce_loss_kernel(const float* __restrict__ z,
                                   float* __restrict__ out) {
    __shared__ float ssum[256];
    __shared__ float smax[256];
    int tid = threadIdx.x;
    float s = 0.f, m = -3.0e38f;
    for (int i = tid; i < N_SEQ; i += 256) {
        float v = z[2 * i] - z[2 * i + 1];
        s += v;
        m = fmaxf(m, v);
    }
    ssum[tid] = s; smax[tid] = m;
    __syncthreads();
    for (int off = 128; off > 0; off >>= 1) {
        if (tid < off) {
            ssum[tid] += ssum[tid + off];
            smax[tid] = fmaxf(smax[tid], smax[tid + off]);
        }
        __syncthreads();
    }
    if (tid == 0) out[0] = (smax[0] > -IMPOSS) ? 0.f : ssum[0] * 0.01f;
}

// ---------------------------------------------------------------------------
extern "C" void kernel_launch(void* const* d_in, const int* in_sizes, int n_in,
                              void* d_out, int out_size, void* d_ws, size_t ws_size,
                              hipStream_t stream) {
    const float*         embeds = (const float*)d_in[0];
    const unsigned char* mask   = (const unsigned char*)d_in[1];
    const int*           spans  = (const int*)d_in[2];
    const float*         W      = (const float*)d_in[3];
    const float*         bias   = (const float*)d_in[4];
    float*               out    = (float*)d_out;

    const int n_spans = in_sizes[2] / 4;

    char* ws = (char*)d_ws;
    const size_t logits_bytes = (size_t)N_SEQ * T_TOK * NUM_TAGS * sizeof(float); // ~21 MB
    const size_t tags_bytes   = (size_t)N_SEQ * T_TOK * sizeof(int);              // 4 MB
    float* logits = (float*)(ws);
    int*   tags   = (int*)(ws + logits_bytes);
    float* z_vals = (float*)(ws + logits_bytes + tags_bytes);                     // 4096 f32

    // K0: clear tags (O = 0)
    hipMemsetAsync(tags, 0, tags_bytes, stream);

    // K1: span scatter
    spans_to_tags_kernel<<<(n_spans + 255) / 256, 256, 0, stream>>>(spans, tags, n_spans);

    // K2: WMMA projection GEMM, writes (n,l,t,tag) layout
    gemm_logits_kernel<<<dim3(N_ROWS / 16), dim3(32, 2), 0, stream>>>(embeds, W, bias, logits);

    // K3: CRF scans, one lane per (sequence, pass)
    crf_scan_kernel<<<(2 * N_SEQ + 255) / 256, 256, 0, stream>>>(logits, tags, mask, z_vals);

    // K4: loss reduction
    reduce_loss_kernel<<<1, 256, 0, stream>>>(z_vals, out);
}